// BidirectionalSAGE_74380243632657
// MI455X (gfx1250) — compile-verified
//
#include <hip/hip_runtime.h>

#define NN 100000
#define EE 1000000
#define DD 64
#define HH 64
#define OUTD 16
#define GG 128

typedef float v2f __attribute__((ext_vector_type(2)));
typedef float v8f __attribute__((ext_vector_type(8)));

// ---------------------------------------------------------------- utilities
__global__ void zero_f32(float* __restrict__ p, int n) {
  int i = blockIdx.x * blockDim.x + threadIdx.x;
  int stride = gridDim.x * blockDim.x;
  for (; i < n; i += stride) p[i] = 0.0f;
}

// degree of each node for forward (dst) and backward (src) aggregation
__global__ void deg_kernel(const int* __restrict__ src, const int* __restrict__ dst,
                           float* __restrict__ degF, float* __restrict__ degB) {
  int e = blockIdx.x * blockDim.x + threadIdx.x;
  if (e >= EE) return;
  atomicAdd(degF + dst[e], 1.0f);
  atomicAdd(degB + src[e], 1.0f);
}

// in-place: deg -> 1/max(deg,1)
__global__ void recip_kernel(float* __restrict__ dF, float* __restrict__ dB) {
  int i = blockIdx.x * blockDim.x + threadIdx.x;
  if (i >= NN) return;
  dF[i] = 1.0f / fmaxf(dF[i], 1.0f);
  dB[i] = 1.0f / fmaxf(dB[i], 1.0f);
}

// 16 lanes per edge; each lane moves 4 features each direction.
// aggF[dst] += X[src];  aggB[src] += X[dst]
__global__ void scatter_kernel(const float* __restrict__ X,
                               const int* __restrict__ src, const int* __restrict__ dst,
                               float* __restrict__ aggF, float* __restrict__ aggB) {
  int tid = blockIdx.x * blockDim.x + threadIdx.x;
  int e = tid >> 4;
  if (e >= EE) return;
  int c = (tid & 15) * 4;
  int s = src[e], d = dst[e];
  float4 xf = *(const float4*)(X + s * DD + c);
  float* pF = aggF + d * DD + c;
  atomicAdd(pF + 0, xf.x);
  atomicAdd(pF + 1, xf.y);
  atomicAdd(pF + 2, xf.z);
  atomicAdd(pF + 3, xf.w);
  float4 xb = *(const float4*)(X + d * DD + c);
  float* pB = aggB + s * DD + c;
  atomicAdd(pB + 0, xb.x);
  atomicAdd(pB + 1, xb.y);
  atomicAdd(pB + 2, xb.z);
  atomicAdd(pB + 3, xb.w);
}

// One wave handles 16 node rows x 64 output cols.
// Y = relu(0.5*( (aggF*rdF)@WlF^T + X@WrF^T + (aggB*rdB)@WlB^T + X@WrB^T + blF + blB ))
// All reads (own 16-row block) complete before stores -> Y may alias aggF.
__global__ __launch_bounds__(256)
void layer_kernel(const float* __restrict__ X,
                  const float* __restrict__ aggF, const float* __restrict__ aggB,
                  const float* __restrict__ rdF, const float* __restrict__ rdB,
                  const float* __restrict__ WlF, const float* __restrict__ blF,
                  const float* __restrict__ WrF,
                  const float* __restrict__ WlB, const float* __restrict__ blB,
                  const float* __restrict__ WrB,
                  float* __restrict__ Y) {
  int wave = blockIdx.x * (blockDim.x >> 5) + (threadIdx.x >> 5);
  int r0 = wave * 16;
  if (r0 >= NN) return;                 // wave-uniform: EXEC all-ones for WMMA
  int lane = threadIdx.x & 31;
  int m  = lane & 15;                   // A: M row / B: N col
  int kh = lane >> 4;                   // K half select
  int arow = r0 + m;
  float rf = rdF[arow];
  float rb = rdB[arow];
  const float* xrow = X    + arow * DD;
  const float* frow = aggF + arow * DD;
  const float* brow = aggB + arow * DD;

  v8f acc[4] = {};                      // 4 N-tiles of 16 cols

  for (int k0 = 0; k0 < DD; k0 += 4) {
    int kk = k0 + 2 * kh;
    v2f aX = *(const v2f*)(xrow + kk);
    v2f aF = (*(const v2f*)(frow + kk)) * rf;
    v2f aB = (*(const v2f*)(brow + kk)) * rb;
#pragma unroll
    for (int nt = 0; nt < 4; ++nt) {
      int n = nt * 16 + m;
      v2f bLF = *(const v2f*)(WlF + n * DD + kk);
      v2f bRF = *(const v2f*)(WrF + n * DD + kk);
      v2f bLB = *(const v2f*)(WlB + n * DD + kk);
      v2f bRB = *(const v2f*)(WrB + n * DD + kk);
      acc[nt] = __builtin_amdgcn_wmma_f32_16x16x4_f32(false, aF, false, bLF, (short)0, acc[nt], false, false);
      acc[nt] = __builtin_amdgcn_wmma_f32_16x16x4_f32(false, aX, false, bRF, (short)0, acc[nt], false, false);
      acc[nt] = __builtin_amdgcn_wmma_f32_16x16x4_f32(false, aB, false, bLB, (short)0, acc[nt], false, false);
      acc[nt] = __builtin_amdgcn_wmma_f32_16x16x4_f32(false, aX, false, bRB, (short)0, acc[nt], false, false);
    }
  }

#pragma unroll
  for (int nt = 0; nt < 4; ++nt) {
    int n = nt * 16 + m;
    float bias = 0.5f * (blF[n] + blB[n]);
#pragma unroll
    for (int v = 0; v < 8; ++v) {
      int row = r0 + v + 8 * kh;
      float val = fmaxf(0.5f * acc[nt][v] + bias, 0.0f);
      Y[row * DD + n] = val;
    }
  }
}

// 16 lanes per node: gsum[batch[node]] += h[node]; gcnt[g] += 1
__global__ void pool_kernel(const float* __restrict__ Hf, const int* __restrict__ batch,
                            float* __restrict__ gsum, float* __restrict__ gcnt) {
  int tid = blockIdx.x * blockDim.x + threadIdx.x;
  int node = tid >> 4;
  if (node >= NN) return;
  int c = (tid & 15) * 4;
  int g = batch[node];
  float4 hv = *(const float4*)(Hf + node * DD + c);
  float* p = gsum + g * DD + c;
  atomicAdd(p + 0, hv.x);
  atomicAdd(p + 1, hv.y);
  atomicAdd(p + 2, hv.z);
  atomicAdd(p + 3, hv.w);
  if ((tid & 15) == 0) atomicAdd(gcnt + g, 1.0f);
}

// out[G,16] = (gsum/cnt) @ pred_W^T + pred_b ; one block, 8 waves (16 graphs each)
__global__ __launch_bounds__(256)
void pred_kernel(const float* __restrict__ gsum, const float* __restrict__ gcnt,
                 const float* __restrict__ W, const float* __restrict__ b,
                 float* __restrict__ out) {
  int wave = threadIdx.x >> 5;
  int g0 = wave * 16;
  int lane = threadIdx.x & 31;
  int m  = lane & 15;
  int kh = lane >> 4;
  int grow = g0 + m;
  float rc = 1.0f / fmaxf(gcnt[grow], 1.0f);
  v8f acc = {};
  for (int k0 = 0; k0 < HH; k0 += 4) {
    int kk = k0 + 2 * kh;
    v2f a = (*(const v2f*)(gsum + grow * HH + kk)) * rc;
    v2f bb = *(const v2f*)(W + m * HH + kk);
    acc = __builtin_amdgcn_wmma_f32_16x16x4_f32(false, a, false, bb, (short)0, acc, false, false);
  }
  float bias = b[m];
#pragma unroll
  for (int v = 0; v < 8; ++v) {
    int row = g0 + v + 8 * kh;
    out[row * OUTD + m] = acc[v] + bias;
  }
}

// ---------------------------------------------------------------- launch
extern "C" void kernel_launch(void* const* d_in, const int* in_sizes, int n_in,
                              void* d_out, int out_size, void* d_ws, size_t ws_size,
                              hipStream_t stream) {
  const float* x      = (const float*)d_in[0];
  const int*   ei     = (const int*)  d_in[1];
  const int*   batch  = (const int*)  d_in[2];
  // d_in[3] = num_graphs (compile-time GG)
  const float* l0f_Wl = (const float*)d_in[4];
  const float* l0f_bl = (const float*)d_in[5];
  const float* l0f_Wr = (const float*)d_in[6];
  const float* l0b_Wl = (const float*)d_in[7];
  const float* l0b_bl = (const float*)d_in[8];
  const float* l0b_Wr = (const float*)d_in[9];
  const float* l1f_Wl = (const float*)d_in[10];
  const float* l1f_bl = (const float*)d_in[11];
  const float* l1f_Wr = (const float*)d_in[12];
  const float* l1b_Wl = (const float*)d_in[13];
  const float* l1b_bl = (const float*)d_in[14];
  const float* l1b_Wr = (const float*)d_in[15];
  const float* pred_W = (const float*)d_in[16];
  const float* pred_b = (const float*)d_in[17];
  float* out = (float*)d_out;

  float* ws   = (float*)d_ws;
  float* A1   = ws;                        // N*64   aggF (layer1 output aliases this)
  float* A2   = A1 + (size_t)NN * DD;      // N*64   aggB
  float* Hbuf = A2 + (size_t)NN * DD;      // N*64   layer0 output
  float* dF   = Hbuf + (size_t)NN * DD;    // N      1/deg forward
  float* dB   = dF + NN;                   // N      1/deg backward
  float* gsum = dB + NN;                   // G*64
  float* gcnt = gsum + (size_t)GG * HH;    // G

  const int* src = ei;
  const int* dst = ei + EE;

  const int TB = 256;
  int layer_blocks = ((NN / 16) + 7) / 8;  // 8 waves per block

  // degrees (shared by both layers)
  zero_f32<<<512, TB, 0, stream>>>(dF, 2 * NN);
  deg_kernel<<<(EE + TB - 1) / TB, TB, 0, stream>>>(src, dst, dF, dB);
  recip_kernel<<<(NN + TB - 1) / TB, TB, 0, stream>>>(dF, dB);

  // ---- layer 0
  zero_f32<<<4096, TB, 0, stream>>>(A1, 2 * NN * DD);
  scatter_kernel<<<(EE * 16) / TB, TB, 0, stream>>>(x, src, dst, A1, A2);
  layer_kernel<<<layer_blocks, TB, 0, stream>>>(x, A1, A2, dF, dB,
                                                l0f_Wl, l0f_bl, l0f_Wr,
                                                l0b_Wl, l0b_bl, l0b_Wr, Hbuf);

  // ---- layer 1 (output aliases A1; safe: each wave reads only its own rows
  //      before storing them)
  zero_f32<<<4096, TB, 0, stream>>>(A1, 2 * NN * DD);
  scatter_kernel<<<(EE * 16) / TB, TB, 0, stream>>>(Hbuf, src, dst, A1, A2);
  layer_kernel<<<layer_blocks, TB, 0, stream>>>(Hbuf, A1, A2, dF, dB,
                                                l1f_Wl, l1f_bl, l1f_Wr,
                                                l1b_Wl, l1b_bl, l1b_Wr, A1);

  // ---- pooling + prediction head
  zero_f32<<<64, TB, 0, stream>>>(gsum, GG * HH + GG);
  pool_kernel<<<(NN * 16 + TB - 1) / TB, TB, 0, stream>>>(A1, batch, gsum, gcnt);
  pred_kernel<<<1, TB, 0, stream>>>(gsum, gcnt, pred_W, pred_b, out);
}